// MolAggregator_55886114456063
// MI455X (gfx1250) — compile-verified
//
#include <hip/hip_runtime.h>
#include <math.h>

// ---------------------------------------------------------------------------
// MI455X (gfx1250) fully fused ASAP-pooling GNN forward.
//
// The batch is 128 independent 32-node graphs (edges e in [g*256,(g+1)*256)
// belong to graph g; self-loops appended). Total work ~1 GFLOP / ~4 MB, i.e.
// launch-latency bound on a 23.3 TB/s part -> fuse the entire forward into a
// single kernel, one 256-thread (8 x wave32) workgroup per graph, all
// intermediates in LDS (~78 KB << 320 KB/WGP), weights L2-resident.
// All GEMMs use V_WMMA_F32_16X16X4_F32 (fp32-precise matrix path).
// ---------------------------------------------------------------------------

typedef __attribute__((ext_vector_type(2))) float v2f;
typedef __attribute__((ext_vector_type(8))) float v8f;

#define GRAPHS 128
#define NPG 32
#define FDIM 128
#define EPG 256                 // edges per graph (E / B)
#define ETOT (GRAPHS * EPG)     // 32768
#define KSEL1 16                // ceil(0.5*32)
#define KSEL2 8                 // ceil(0.5*16)
#define OUTSTR (5 * FDIM)       // [mol | r2(mean,max) | r1(mean,max)]
#define NEG_HUGE (-3.402823466e38f)

#if __has_builtin(__builtin_amdgcn_wmma_f32_16x16x4_f32)
#define HAVE_WMMA_F32X4 1
#endif

static __device__ __forceinline__ v8f wmma_f32_4(v2f a, v2f b, v8f c) {
#ifdef HAVE_WMMA_F32X4
  // (neg_a, A, neg_b, B, c_mod, C, reuse_a, reuse_b)
  return __builtin_amdgcn_wmma_f32_16x16x4_f32(false, a, false, b, (short)0, c,
                                               false, false);
#else
  c[0] += a.x * b.x + a.y * b.y;  // keep-compiling placeholder
  return c;
#endif
}

// D(16x16) += A(16xK) * B(Kx16).  A addressed A[m*ars + k*acs],
// B addressed B[k*brs + n*bcs]; works for row-major / transposed / LDS /
// global operands.  Must be called wave-uniformly (EXEC all ones).
// Frag layout per ISA 7.12.2: A f32 16x4: lane m<16 holds (K=k+0,k+1) in the
// two VGPRs, lane m+16 holds (K=k+2,k+3).  B 4x16: rows striped across lanes.
static __device__ __forceinline__ v8f mm_tile(const float* A, int ars, int acs,
                                              int m0, const float* B, int brs,
                                              int bcs, int n0, int K, v8f acc,
                                              int lane) {
  const int m = m0 + (lane & 15);
  const int kka = (lane >> 4) << 1;  // 0 / 2
  const int n = n0 + (lane & 15);
  const int krb = (lane >> 4);  // 0 / 1
  for (int k = 0; k < K; k += 4) {
    v2f a, b;
    a.x = A[m * ars + (k + kka) * acs];
    a.y = A[m * ars + (k + kka + 1) * acs];
    b.x = B[(k + krb) * brs + n * bcs];
    b.y = B[(k + krb + 2) * brs + n * bcs];
    acc = wmma_f32_4(a, b, acc);
  }
  return acc;
}

// C/D layout: VGPR r, lanes 0-15 -> row m0+r, lanes 16-31 -> row m0+r+8.
static __device__ __forceinline__ void store_tile(float* D, int ldd, int m0,
                                                  int n0, v8f acc, int lane) {
  const int col = n0 + (lane & 15);
  const int mb = m0 + ((lane >> 4) << 3);
#pragma unroll
  for (int r = 0; r < 8; ++r) D[(mb + r) * ldd + col] = acc[r];
}

static __device__ __forceinline__ float lrelu(float v) {
  return v > 0.f ? v : 0.2f * v;
}
static __device__ __forceinline__ float sigmoidf_(float v) {
  return 1.f / (1.f + expf(-v));
}

__global__ __launch_bounds__(256) void molagg_fused_kernel(
    const float* __restrict__ x, const float* __restrict__ p1_rel_w,
    const float* __restrict__ p1_rel_b, const float* __restrict__ p1_root_w,
    const float* __restrict__ p1_lin_w, const float* __restrict__ p1_lin_b,
    const float* __restrict__ p1_att_w, const float* __restrict__ p1_att_b,
    const float* __restrict__ p1_le1_w, const float* __restrict__ p1_le1_b,
    const float* __restrict__ p1_le2_w, const float* __restrict__ p1_le3_w,
    const float* __restrict__ p1_le3_b, const float* __restrict__ p2_rel_w,
    const float* __restrict__ p2_rel_b, const float* __restrict__ p2_root_w,
    const float* __restrict__ p2_lin_w, const float* __restrict__ p2_lin_b,
    const float* __restrict__ p2_att_w, const float* __restrict__ p2_att_b,
    const float* __restrict__ p2_le1_w, const float* __restrict__ p2_le1_b,
    const float* __restrict__ p2_le2_w, const float* __restrict__ p2_le3_w,
    const float* __restrict__ p2_le3_b, const float* __restrict__ conv1_w,
    const float* __restrict__ conv2_w, const float* __restrict__ ga_gate_w,
    const float* __restrict__ ga_gate_b, const float* __restrict__ ga_nn_w,
    const float* __restrict__ ga_nn_b, const int* __restrict__ edge_index,
    float* __restrict__ out) {
  __shared__ float S0[NPG * FDIM];  // x -> T1 scratch -> hh/h1n -> h2 -> u
  __shared__ float S1[NPG * FDIM];  // agg -> qpre -> xc -> hmix -> t2 -> xc2..
  __shared__ float S2[NPG * FDIM];  // x_pool -> h1 -> x_pool2 -> x2
  __shared__ float S3[NPG * FDIM];  // x_q -> x1 (lo) + xq2 (hi) -> h2n
  __shared__ float sA[NPG * NPG];   // level-0 adjacency counts (w/ self loops)
  __shared__ float sS[NPG * NPG];   // level-0 score matrix
  __shared__ float sM[KSEL1 * KSEL1];   // A2 (binary)
  __shared__ float sM2[KSEL1 * KSEL1];  // An^T / T3 / An3^T scratch
  __shared__ float sS2[KSEL1 * KSEL1];  // level-1 score matrix
  __shared__ float sA3[KSEL1 * KSEL1];  // A3 (binary, zero-padded to 16x16)
  __shared__ float sv0[NPG], sv1[NPG], sv2[NPG], sv3[NPG];
  __shared__ int sperm[KSEL1];
  __shared__ int sperm2[KSEL2];

  const int tid = threadIdx.x;
  const int lane = tid & 31;
  const int wid = tid >> 5;
  const int g = blockIdx.x;
  float* outg = out + (size_t)g * OUTSTR;
  const v8f z8 = {0.f, 0.f, 0.f, 0.f, 0.f, 0.f, 0.f, 0.f};

  // ---- Stage 0: stage x into LDS, zero adjacency --------------------------
  {
    const float* xg = x + (size_t)g * NPG * FDIM;
    for (int i = tid; i < NPG * FDIM; i += 256) S0[i] = xg[i];
    for (int i = tid; i < NPG * NPG; i += 256) sA[i] = 0.f;
  }
  __syncthreads();

  // ---- Stage 1: r1 readout + adjacency counts -----------------------------
  if (tid < FDIM) {
    float s = 0.f, m = NEG_HUGE;
    for (int i = 0; i < NPG; ++i) {
      float v = S0[i * FDIM + tid];
      s += v;
      m = fmaxf(m, v);
    }
    outg[3 * FDIM + tid] = s * (1.f / NPG);
    outg[4 * FDIM + tid] = m;
  }
  {
    const int e = g * EPG + tid;  // one edge per thread (EPG == blockDim)
    const int sl = edge_index[e] - g * NPG;
    const int dl = edge_index[ETOT + e] - g * NPG;
    atomicAdd(&sA[sl * NPG + dl], 1.f);
  }
  __syncthreads();
  if (tid < NPG) sA[tid * NPG + tid] += 1.f;  // self loops
  __syncthreads();

  // ---- Stage 2: agg = A^T x (WMMA, K=32) -> S1 ----------------------------
  for (int t = wid; t < 16; t += 8) {
    const int m0 = (t >> 3) << 4, n0 = (t & 7) << 4;
    v8f acc = z8;
    acc = mm_tile(sA, 1, NPG, m0, S0, FDIM, 1, n0, NPG, acc, lane);
    store_tile(S1, FDIM, m0, n0, acc, lane);
  }
  __syncthreads();

  // ---- Stage 3: x_pool = agg@rel_w + x@root_w + rel_b -> S2 ---------------
  for (int t = wid; t < 16; t += 8) {
    const int m0 = (t >> 3) << 4, n0 = (t & 7) << 4;
    v8f acc = z8;
    acc = mm_tile(S1, FDIM, 1, m0, p1_rel_w, FDIM, 1, n0, FDIM, acc, lane);
    acc = mm_tile(S0, FDIM, 1, m0, p1_root_w, FDIM, 1, n0, FDIM, acc, lane);
    const int col = n0 + (lane & 15);
    const int mb = m0 + ((lane >> 4) << 3);
    const float bb = p1_rel_b[col];
#pragma unroll
    for (int r = 0; r < 8; ++r) S2[(mb + r) * FDIM + col] = acc[r] + bb;
  }
  __syncthreads();

  // ---- Stage 4: x_q_pre[j] = max_{i:A[i,j]>0} x_pool[i] -> S1 -------------
  {
    const int j = tid >> 3, f0 = (tid & 7) << 4;
    float mx[16];
#pragma unroll
    for (int r = 0; r < 16; ++r) mx[r] = NEG_HUGE;
    for (int i = 0; i < NPG; ++i)
      if (sA[i * NPG + j] > 0.f) {
#pragma unroll
        for (int r = 0; r < 16; ++r)
          mx[r] = fmaxf(mx[r], S2[i * FDIM + f0 + r]);
      }
#pragma unroll
    for (int r = 0; r < 16; ++r) S1[j * FDIM + f0 + r] = mx[r];
  }
  __syncthreads();

  // ---- Stage 5: x_q = x_q_pre@lin_w + lin_b -> S3 -------------------------
  for (int t = wid; t < 16; t += 8) {
    const int m0 = (t >> 3) << 4, n0 = (t & 7) << 4;
    v8f acc = z8;
    acc = mm_tile(S1, FDIM, 1, m0, p1_lin_w, FDIM, 1, n0, FDIM, acc, lane);
    const int col = n0 + (lane & 15);
    const int mb = m0 + ((lane >> 4) << 3);
    const float bb = p1_lin_b[col];
#pragma unroll
    for (int r = 0; r < 8; ++r) S3[(mb + r) * FDIM + col] = acc[r] + bb;
  }
  __syncthreads();

  // ---- Stage 6: per-node attention halves ---------------------------------
  if (tid < 64) {
    const int i = tid & 31;
    const float* row = (tid < 32) ? (S2 + i * FDIM) : (S3 + i * FDIM);
    const float* w = (tid < 32) ? (p1_att_w + FDIM) : p1_att_w;
    float s = 0.f;
    for (int f = 0; f < FDIM; ++f) s += row[f] * w[f];
    if (tid < 32)
      sv0[i] = s;  // ap = x_pool . att_w[F:]
    else
      sv1[i] = s;  // aq = x_q    . att_w[:F]
  }
  __syncthreads();

  // ---- Stage 7: multiplicity-weighted masked softmax per column -> sS -----
  if (tid < NPG) {
    const int j = tid;
    const float ab = p1_att_b[0], aq = sv1[j];
    float mx = NEG_HUGE;
    for (int i = 0; i < NPG; ++i)
      if (sA[i * NPG + j] > 0.f) mx = fmaxf(mx, lrelu(aq + sv0[i] + ab));
    float den = 0.f;
    for (int i = 0; i < NPG; ++i) {
      const float c = sA[i * NPG + j];
      if (c > 0.f) den += c * expf(lrelu(aq + sv0[i] + ab) - mx);
    }
    const float inv = 1.f / den;
    for (int i = 0; i < NPG; ++i) {
      const float c = sA[i * NPG + j];
      sS[i * NPG + j] =
          (c > 0.f) ? c * expf(lrelu(aq + sv0[i] + ab) - mx) * inv : 0.f;
    }
  }
  __syncthreads();

  // ---- Stage 8: xc = S^T x (WMMA, K=32) -> S1 -----------------------------
  for (int t = wid; t < 16; t += 8) {
    const int m0 = (t >> 3) << 4, n0 = (t & 7) << 4;
    v8f acc = z8;
    acc = mm_tile(sS, 1, NPG, m0, S0, FDIM, 1, n0, NPG, acc, lane);
    store_tile(S1, FDIM, m0, n0, acc, lane);
  }
  __syncthreads();

  // ---- Stage 9: fitness ---------------------------------------------------
  if (tid < NPG) {
    const float* row = S1 + tid * FDIM;
    float s1 = 0.f, s2 = 0.f, s3 = 0.f;
    for (int f = 0; f < FDIM; ++f) {
      const float v = row[f];
      s1 += v * p1_le1_w[f];
      s2 += v * p1_le2_w[f];
      s3 += v * p1_le3_w[f];
    }
    sv0[tid] = s1 + p1_le1_b[0];
    sv1[tid] = s2;
    sv2[tid] = s3 + p1_le3_b[0];
  }
  __syncthreads();
  if (tid < NPG) {
    float acc = 0.f, deg = 0.f;
    for (int i = 0; i < NPG; ++i) {
      const float c = sA[i * NPG + tid];
      acc += c * sv0[i];
      deg += c;
    }
    sv3[tid] = sigmoidf_(acc - deg * sv1[tid] + sv2[tid]);
  }
  __syncthreads();

  // ---- Stage 10: stable top-16 (= argsort(-fit) order) --------------------
  if (tid < NPG) {
    const float fi = sv3[tid];
    int r = 0;
    for (int j = 0; j < NPG; ++j) {
      const float fj = sv3[j];
      r += (fj > fi) || (fj == fi && j < tid);
    }
    if (r < KSEL1) sperm[r] = tid;
  }
  __syncthreads();

  // ---- Stage 11: h1=(xc*fit)[perm] -> S2 ; x1=x[perm] -> S3 ---------------
  for (int idx = tid; idx < KSEL1 * FDIM; idx += 256) {
    const int rr = idx >> 7, f = idx & (FDIM - 1);
    const int p = sperm[rr];
    S2[rr * FDIM + f] = S1[p * FDIM + f] * sv3[p];
    S3[rr * FDIM + f] = S0[p * FDIM + f];
  }
  __syncthreads();

  // ---- Stage 12: A2 = bin(Ssel^T (A Ssel)), diag=1 ------------------------
  for (int idx = tid; idx < NPG * KSEL1; idx += 256) {
    const int i = idx >> 4, c = idx & 15;
    const int pc = sperm[c];
    float s = 0.f;
    for (int k = 0; k < NPG; ++k) s += sA[i * NPG + k] * sS[k * NPG + pc];
    S0[i * KSEL1 + c] = s;  // T1 scratch (x is dead now)
  }
  __syncthreads();
  {
    const int r = tid >> 4, c = tid & 15;
    const int pr = sperm[r];
    float s = 0.f;
    for (int i = 0; i < NPG; ++i) s += sS[i * NPG + pr] * S0[i * KSEL1 + c];
    sM[r * KSEL1 + c] = (r == c) ? 1.f : ((s != 0.f) ? 1.f : 0.f);
  }
  __syncthreads();

  // ---- Stage 13: GCN2 layer 1 + instance norm + relu + r2 -----------------
  if (tid < KSEL1) {
    float s = 0.f;
    for (int r = 0; r < KSEL1; ++r) s += sM[r * KSEL1 + tid];
    sv0[tid] = rsqrtf(s);  // dinv (col degree >= 1)
  }
  __syncthreads();
  {
    const int m = tid >> 4, k = tid & 15;  // An^T row-major
    sM2[m * KSEL1 + k] = sv0[k] * sv0[m] * sM[k * KSEL1 + m];
  }
  __syncthreads();
  {  // hmix = 0.8*(An^T h1) + 0.2*x1 -> S1
    const int n0 = wid << 4;
    v8f acc = z8;
    acc = mm_tile(sM2, KSEL1, 1, 0, S2, FDIM, 1, n0, KSEL1, acc, lane);
    const int col = n0 + (lane & 15);
    const int mb = (lane >> 4) << 3;
#pragma unroll
    for (int r = 0; r < 8; ++r) {
      const int row = mb + r;
      S1[row * FDIM + col] = 0.8f * acc[r] + 0.2f * S3[row * FDIM + col];
    }
  }
  __syncthreads();
  {  // hh = hmix @ conv1_w -> S0
    const int n0 = wid << 4;
    v8f acc = z8;
    acc = mm_tile(S1, FDIM, 1, 0, conv1_w, FDIM, 1, n0, FDIM, acc, lane);
    store_tile(S0, FDIM, 0, n0, acc, lane);
  }
  __syncthreads();
  if (tid < FDIM) {
    float mean = 0.f;
    for (int r = 0; r < KSEL1; ++r) mean += S0[r * FDIM + tid];
    mean *= (1.f / KSEL1);
    float var = 0.f;
    for (int r = 0; r < KSEL1; ++r) {
      const float d = S0[r * FDIM + tid] - mean;
      var += d * d;
    }
    var *= (1.f / KSEL1);
    const float inv = rsqrtf(var + 1e-5f);
    float s = 0.f, mx = NEG_HUGE;
    for (int r = 0; r < KSEL1; ++r) {
      float h = (S0[r * FDIM + tid] - mean) * inv;
      h = fmaxf(h, 0.f);
      S0[r * FDIM + tid] = h;  // h1n
      s += h;
      mx = fmaxf(mx, h);
    }
    outg[FDIM + tid] = s * (1.f / KSEL1);
    outg[2 * FDIM + tid] = mx;
  }
  __syncthreads();

  // ---- Stage 14: ASAP level 2 ---------------------------------------------
  float* XQ2 = S3 + KSEL1 * FDIM;  // rows 16..31 of S3 are free

  {  // t2 = A2^T h1n -> S1
    const int n0 = wid << 4;
    v8f acc = z8;
    acc = mm_tile(sM, 1, KSEL1, 0, S0, FDIM, 1, n0, KSEL1, acc, lane);
    store_tile(S1, FDIM, 0, n0, acc, lane);
  }
  __syncthreads();
  {  // x_pool2 = t2@rel2 + h1n@root2 + rel2_b -> S2
    const int n0 = wid << 4;
    v8f acc = z8;
    acc = mm_tile(S1, FDIM, 1, 0, p2_rel_w, FDIM, 1, n0, FDIM, acc, lane);
    acc = mm_tile(S0, FDIM, 1, 0, p2_root_w, FDIM, 1, n0, FDIM, acc, lane);
    const int col = n0 + (lane & 15);
    const int mb = (lane >> 4) << 3;
    const float bb = p2_rel_b[col];
#pragma unroll
    for (int r = 0; r < 8; ++r) S2[(mb + r) * FDIM + col] = acc[r] + bb;
  }
  __syncthreads();
  if (tid < 128) {  // qpre2 -> S1
    const int j = tid >> 3, f0 = (tid & 7) << 4;
    float mx[16];
#pragma unroll
    for (int r = 0; r < 16; ++r) mx[r] = NEG_HUGE;
    for (int i = 0; i < KSEL1; ++i)
      if (sM[i * KSEL1 + j] > 0.f) {
#pragma unroll
        for (int r = 0; r < 16; ++r)
          mx[r] = fmaxf(mx[r], S2[i * FDIM + f0 + r]);
      }
#pragma unroll
    for (int r = 0; r < 16; ++r) S1[j * FDIM + f0 + r] = mx[r];
  }
  __syncthreads();
  {  // xq2 = qpre2@lin2 + b -> XQ2
    const int n0 = wid << 4;
    v8f acc = z8;
    acc = mm_tile(S1, FDIM, 1, 0, p2_lin_w, FDIM, 1, n0, FDIM, acc, lane);
    const int col = n0 + (lane & 15);
    const int mb = (lane >> 4) << 3;
    const float bb = p2_lin_b[col];
#pragma unroll
    for (int r = 0; r < 8; ++r) XQ2[(mb + r) * FDIM + col] = acc[r] + bb;
  }
  __syncthreads();
  if (tid < 2 * KSEL1) {  // attention halves
    const int i = tid & 15;
    const float* row = (tid < KSEL1) ? (S2 + i * FDIM) : (XQ2 + i * FDIM);
    const float* w = (tid < KSEL1) ? (p2_att_w + FDIM) : p2_att_w;
    float s = 0.f;
    for (int f = 0; f < FDIM; ++f) s += row[f] * w[f];
    if (tid < KSEL1)
      sv0[i] = s;
    else
      sv1[i] = s;
  }
  __syncthreads();
  if (tid < KSEL1) {  // masked softmax -> sS2
    const int j = tid;
    const float ab = p2_att_b[0], aq = sv1[j];
    float mx = NEG_HUGE;
    for (int i = 0; i < KSEL1; ++i)
      if (sM[i * KSEL1 + j] > 0.f) mx = fmaxf(mx, lrelu(aq + sv0[i] + ab));
    float den = 0.f;
    for (int i = 0; i < KSEL1; ++i)
      if (sM[i * KSEL1 + j] > 0.f) den += expf(lrelu(aq + sv0[i] + ab) - mx);
    const float inv = 1.f / den;
    for (int i = 0; i < KSEL1; ++i)
      sS2[i * KSEL1 + j] = (sM[i * KSEL1 + j] > 0.f)
                               ? expf(lrelu(aq + sv0[i] + ab) - mx) * inv
                               : 0.f;
  }
  __syncthreads();
  {  // xc2 = S2^T h1n -> S1
    const int n0 = wid << 4;
    v8f acc = z8;
    acc = mm_tile(sS2, 1, KSEL1, 0, S0, FDIM, 1, n0, KSEL1, acc, lane);
    store_tile(S1, FDIM, 0, n0, acc, lane);
  }
  __syncthreads();
  if (tid < KSEL1) {  // fitness 2
    const float* row = S1 + tid * FDIM;
    float s1 = 0.f, s2 = 0.f, s3 = 0.f;
    for (int f = 0; f < FDIM; ++f) {
      const float v = row[f];
      s1 += v * p2_le1_w[f];
      s2 += v * p2_le2_w[f];
      s3 += v * p2_le3_w[f];
    }
    sv0[tid] = s1 + p2_le1_b[0];
    sv1[tid] = s2;
    sv2[tid] = s3 + p2_le3_b[0];
  }
  __syncthreads();
  if (tid < KSEL1) {
    float acc = 0.f, deg = 0.f;
    for (int i = 0; i < KSEL1; ++i) {
      const float c = sM[i * KSEL1 + tid];
      acc += c * sv0[i];
      deg += c;
    }
    sv3[tid] = sigmoidf_(acc - deg * sv1[tid] + sv2[tid]);
  }
  __syncthreads();
  if (tid < KSEL1) {  // stable top-8
    const float fi = sv3[tid];
    int r = 0;
    for (int j = 0; j < KSEL1; ++j) {
      const float fj = sv3[j];
      r += (fj > fi) || (fj == fi && j < tid);
    }
    if (r < KSEL2) sperm2[r] = tid;
  }
  __syncthreads();
  // gather h2 -> S0 (zero-padded to 16 rows), x2 -> S2 (zero-padded)
  for (int idx = tid; idx < KSEL1 * FDIM; idx += 256) {
    const int rr = idx >> 7, f = idx & (FDIM - 1);
    float hv = 0.f, xv = 0.f;
    if (rr < KSEL2) {
      const int p = sperm2[rr];
      hv = S1[p * FDIM + f] * sv3[p];
      xv = S3[p * FDIM + f];
    }
    S0[rr * FDIM + f] = hv;
    S2[rr * FDIM + f] = xv;
  }
  __syncthreads();
  // T3 = A2 @ Ssel2 (16x8) -> sM2
  for (int idx = tid; idx < KSEL1 * KSEL2; idx += 256) {
    const int i = idx >> 3, c = idx & 7;
    const int pc = sperm2[c];
    float s = 0.f;
    for (int k = 0; k < KSEL1; ++k)
      s += sM[i * KSEL1 + k] * sS2[k * KSEL1 + pc];
    sM2[i * KSEL2 + c] = s;
  }
  __syncthreads();
  {  // A3 = bin(Ssel2^T T3), diag=1, zero-padded -> sA3
    const int r = tid >> 4, c = tid & 15;
    float v = 0.f;
    if (r < KSEL2 && c < KSEL2) {
      const int pr = sperm2[r];
      float s = 0.f;
      for (int i = 0; i < KSEL1; ++i)
        s += sS2[i * KSEL1 + pr] * sM2[i * KSEL2 + c];
      v = (r == c) ? 1.f : ((s != 0.f) ? 1.f : 0.f);
    }
    sA3[r * KSEL1 + c] = v;
  }
  __syncthreads();

  // ---- Stage 15: GCN2 layer 2 + instance norm + relu ----------------------
  if (tid < KSEL2) {
    float s = 0.f;
    for (int r = 0; r < KSEL2; ++r) s += sA3[r * KSEL1 + tid];
    sv0[tid] = rsqrtf(s);
  }
  __syncthreads();
  {
    const int m = tid >> 4, k = tid & 15;  // An3^T, zero-padded
    sM2[m * KSEL1 + k] = (m < KSEL2 && k < KSEL2)
                             ? sv0[k] * sv0[m] * sA3[k * KSEL1 + m]
                             : 0.f;
  }
  __syncthreads();
  {  // hmix2 = 0.8*(An3^T h2) + 0.2*x2 -> S1
    const int n0 = wid << 4;
    v8f acc = z8;
    acc = mm_tile(sM2, KSEL1, 1, 0, S0, FDIM, 1, n0, KSEL1, acc, lane);
    const int col = n0 + (lane & 15);
    const int mb = (lane >> 4) << 3;
#pragma unroll
    for (int r = 0; r < 8; ++r) {
      const int row = mb + r;
      S1[row * FDIM + col] = 0.8f * acc[r] + 0.2f * S2[row * FDIM + col];
    }
  }
  __syncthreads();
  {  // hh2 = hmix2 @ conv2_w -> S3 (rows 8..15 stay zero)
    const int n0 = wid << 4;
    v8f acc = z8;
    acc = mm_tile(S1, FDIM, 1, 0, conv2_w, FDIM, 1, n0, FDIM, acc, lane);
    store_tile(S3, FDIM, 0, n0, acc, lane);
  }
  __syncthreads();
  if (tid < FDIM) {
    float mean = 0.f;
    for (int r = 0; r < KSEL2; ++r) mean += S3[r * FDIM + tid];
    mean *= (1.f / KSEL2);
    float var = 0.f;
    for (int r = 0; r < KSEL2; ++r) {
      const float d = S3[r * FDIM + tid] - mean;
      var += d * d;
    }
    var *= (1.f / KSEL2);
    const float inv = rsqrtf(var + 1e-5f);
    for (int r = 0; r < KSEL2; ++r) {
      const float h = (S3[r * FDIM + tid] - mean) * inv;
      S3[r * FDIM + tid] = fmaxf(h, 0.f);  // h2n
    }
  }
  __syncthreads();

  // ---- Stage 16: gated global-attention readout ---------------------------
  if (tid < KSEL2) {
    const float* row = S3 + tid * FDIM;
    float s = 0.f;
    for (int f = 0; f < FDIM; ++f) s += row[f] * ga_gate_w[f];
    sv1[tid] = s + ga_gate_b[0];
  }
  __syncthreads();
  if (tid == 0) {
    float mx = NEG_HUGE;
    for (int j = 0; j < KSEL2; ++j) mx = fmaxf(mx, sv1[j]);
    float den = 0.f;
    for (int j = 0; j < KSEL2; ++j) den += expf(sv1[j] - mx);
    const float inv = 1.f / den;
    for (int j = 0; j < KSEL2; ++j) sv2[j] = expf(sv1[j] - mx) * inv;
  }
  __syncthreads();
  {  // u = h2n @ ga_nn_w + ga_nn_b -> S0
    const int n0 = wid << 4;
    v8f acc = z8;
    acc = mm_tile(S3, FDIM, 1, 0, ga_nn_w, FDIM, 1, n0, FDIM, acc, lane);
    const int col = n0 + (lane & 15);
    const int mb = (lane >> 4) << 3;
    const float bb = ga_nn_b[col];
#pragma unroll
    for (int r = 0; r < 8; ++r) S0[(mb + r) * FDIM + col] = acc[r] + bb;
  }
  __syncthreads();
  if (tid < FDIM) {
    float s = 0.f;
    for (int j = 0; j < KSEL2; ++j) s += sv2[j] * S0[j * FDIM + tid];
    outg[tid] = s;  // mol
  }
}

extern "C" void kernel_launch(void* const* d_in, const int* in_sizes, int n_in,
                              void* d_out, int out_size, void* d_ws,
                              size_t ws_size, hipStream_t stream) {
  (void)in_sizes;
  (void)n_in;
  (void)d_ws;
  (void)ws_size;
  (void)out_size;
  molagg_fused_kernel<<<dim3(GRAPHS), dim3(256), 0, stream>>>(
      (const float*)d_in[0], (const float*)d_in[1], (const float*)d_in[2],
      (const float*)d_in[3], (const float*)d_in[4], (const float*)d_in[5],
      (const float*)d_in[6], (const float*)d_in[7], (const float*)d_in[8],
      (const float*)d_in[9], (const float*)d_in[10], (const float*)d_in[11],
      (const float*)d_in[12], (const float*)d_in[13], (const float*)d_in[14],
      (const float*)d_in[15], (const float*)d_in[16], (const float*)d_in[17],
      (const float*)d_in[18], (const float*)d_in[19], (const float*)d_in[20],
      (const float*)d_in[21], (const float*)d_in[22], (const float*)d_in[23],
      (const float*)d_in[24], (const float*)d_in[25], (const float*)d_in[26],
      (const float*)d_in[27], (const float*)d_in[28], (const float*)d_in[29],
      (const float*)d_in[30], (const int*)d_in[31], (float*)d_out);
}